// GraphConnectionsNN_74397423501329
// MI455X (gfx1250) — compile-verified
//
#include <hip/hip_runtime.h>

// ---------------------------------------------------------------------------
// GraphConnectionsNN on MI455X (gfx1250, wave32)
//
// Reference collapses to:
//   x[n]  = lrelu( sum_k model_input[idx[n,k]] * W1[n,k] + b1[n] )   n<500000
//   y2    = lrelu( x @ W2 + b2 )        W2: [500000, 256]  (512 MB -> dominant)
//   y3    = lrelu( y2 @ W3 + b3 )       [256,128]
//   out   = lrelu( y3 @ W4 + b4 )       [128,1] -> scalar
//
// Bandwidth-bound: ~640 MB total traffic, ~0.5 GFLOP. Strategy: fuse step 1
// with the W2 GEMV so x never hits HBM; per-block partial sums reduced in a
// fixed order (deterministic, no float atomics). The GEMV uses
// V_WMMA_F32_16X16X4_F32 with M padded 1->16 by replicating x across A rows.
// ---------------------------------------------------------------------------

typedef __attribute__((ext_vector_type(2))) float v2f;
typedef __attribute__((ext_vector_type(8))) float v8f;

#define NN       500000
#define KK       32
#define H1       256
#define H2       128
#define CHUNK    500          // nodes per block; 500000 = 1000 * 500; 500 % 4 == 0
#define NBLK     (NN / CHUNK) // 1000
#define NEG_SLOPE 0.01f

__device__ __forceinline__ float lrelu(float v) {
    return v > 0.0f ? v : NEG_SLOPE * v;
}

// ---------------------------------------------------------------------------
// Kernel A: per block, compute x for CHUNK nodes (phase 1), then accumulate
// the block's contribution to y2 = x @ W2 with f32 WMMA (phase 2).
// 512 threads = 16 waves. Wave w owns output columns [16w, 16w+16).
// ---------------------------------------------------------------------------
__global__ void __launch_bounds__(512)
gcnn_fused_gather_gemv(const float* __restrict__ mi,    // [N] model_input
                       const int*   __restrict__ idx,   // [N,K]
                       const float* __restrict__ W1,    // [N,K]
                       const float* __restrict__ b1,    // [N]
                       const float* __restrict__ W2,    // [N,H1]
                       float*       __restrict__ partials) // [NBLK, H1]
{
    __shared__ float xs[CHUNK];

    const int tid  = threadIdx.x;
    const int lane = tid & 31;
    const int wv   = tid >> 5;          // wave id, 0..15
    const int base = blockIdx.x * CHUNK;

    // ---------------- Phase 1: one wave per node, coalesced 128B loads -----
    for (int nl = wv; nl < CHUNK; nl += 16) {
        const int n = base + nl;
        const int off = n * KK + lane;
        const int   j = idx[off];       // coalesced across 32 lanes
        const float w = W1[off];        // coalesced
        float p = mi[j] * w;            // gather: 2MB table, L2-resident
        // wave32 butterfly reduction
        #pragma unroll
        for (int s = 16; s > 0; s >>= 1)
            p += __shfl_xor(p, s, 32);
        if (lane == 0)
            xs[nl] = lrelu(p + b1[n]);
    }
    __syncthreads();

    // ---------------- Phase 2: WMMA GEMV over the chunk --------------------
    // A (16x4 f32): all 16 rows replicated with x[i..i+3].
    //   lanes 0-15 : VGPR0=x[i],   VGPR1=x[i+1]
    //   lanes 16-31: VGPR0=x[i+2], VGPR1=x[i+3]
    // B (4x16 f32): B[k][n] = W2[row i+k, col 16*wv + n].
    //   lanes 0-15 : VGPR0=row i   cols, VGPR1=row i+1 cols
    //   lanes 16-31: VGPR0=row i+2 cols, VGPR1=row i+3 cols
    // D (16x16 f32): row M=0 lives in VGPR0 of lanes 0-15 (N=lane).
    const int col   = (wv << 4) + (lane & 15);
    const int khalf = (lane >> 4) << 1;          // 0 for lanes 0-15, 2 for 16-31

    v8f c = {};
    for (int i = 0; i < CHUNK; i += 4) {
        v2f a, b;
        a.x = xs[i + khalf];
        a.y = xs[i + khalf + 1];
        const long long row = (long long)(base + i + khalf);
        b.x = W2[row * H1 + col];
        b.y = W2[(row + 1) * H1 + col];
        c = __builtin_amdgcn_wmma_f32_16x16x4_f32(
                /*neg_a=*/false, a, /*neg_b=*/false, b,
                /*c_mod=*/(short)0, c, /*reuse_a=*/false, /*reuse_b=*/false);
    }

    if (lane < 16)
        partials[blockIdx.x * H1 + (wv << 4) + lane] = c[0];  // M=0 row
}

// ---------------------------------------------------------------------------
// Kernel B: deterministic fixed-order reduction of block partials, then the
// tiny 256->128->1 tail. One block, 256 threads.
// ---------------------------------------------------------------------------
__global__ void __launch_bounds__(256)
gcnn_tail(const float* __restrict__ partials, // [NBLK, H1]
          const float* __restrict__ b2,       // [H1]
          const float* __restrict__ W3,       // [H1,H2]
          const float* __restrict__ b3,       // [H2]
          const float* __restrict__ W4,       // [H2,1]
          const float* __restrict__ b4,       // [1]
          float*       __restrict__ out)      // [1]
{
    __shared__ float y2s[H1];
    __shared__ float h2s[H2];

    const int t = threadIdx.x;

    // y2[t] = lrelu( sum_blocks partials[c][t] + b2[t] )  (fixed order)
    float s = 0.0f;
    for (int cidx = 0; cidx < NBLK; ++cidx)
        s += partials[cidx * H1 + t];          // coalesced across threads
    y2s[t] = lrelu(s + b2[t]);
    __syncthreads();

    // y3[j] = lrelu( sum_k y2[k] * W3[k][j] + b3[j] )
    if (t < H2) {
        float h = 0.0f;
        for (int k = 0; k < H1; ++k)
            h += y2s[k] * W3[k * H2 + t];
        h2s[t] = lrelu(h + b3[t]);
    }
    __syncthreads();

    // out = lrelu( sum_j y3[j] * W4[j] + b4 )
    if (t == 0) {
        float o = 0.0f;
        for (int j = 0; j < H2; ++j)
            o += h2s[j] * W4[j];
        out[0] = lrelu(o + b4[0]);
    }
}

// ---------------------------------------------------------------------------
extern "C" void kernel_launch(void* const* d_in, const int* in_sizes, int n_in,
                              void* d_out, int out_size, void* d_ws, size_t ws_size,
                              hipStream_t stream) {
    const float* mi  = (const float*)d_in[0];  // model_input [N,1]
    const int*   idx = (const int*)  d_in[1];  // [N,K]
    const float* W1  = (const float*)d_in[2];  // [N,K]
    const float* b1  = (const float*)d_in[3];  // [N]
    const float* W2  = (const float*)d_in[4];  // [N,H1]
    const float* b2  = (const float*)d_in[5];  // [H1]
    const float* W3  = (const float*)d_in[6];  // [H1,H2]
    const float* b3  = (const float*)d_in[7];  // [H2]
    const float* W4  = (const float*)d_in[8];  // [H2,1]
    const float* b4  = (const float*)d_in[9];  // [1]
    float* out      = (float*)d_out;
    float* partials = (float*)d_ws;            // NBLK * H1 * 4 = 1.0 MB

    gcnn_fused_gather_gemv<<<NBLK, 512, 0, stream>>>(mi, idx, W1, b1, W2, partials);
    gcnn_tail<<<1, 256, 0, stream>>>(partials, b2, W3, b3, W4, b4, out);
}